// AdaptiveCenterLoss_56521769615881
// MI455X (gfx1250) — compile-verified
//
#include <hip/hip_runtime.h>

// AdaptiveCenterLoss, algebraically reduced:
//   loss = (1/B) * sum_b relu( ||x_b - centers[labels_b]||^2 - margin )
// The one-hot mask kills the full B x C GEMM; only B=1024 gathered rows matter.
// Per 16-row tile we form S = X - G and compute diag(S * S^T) with chained
// V_WMMA_F32_16X16X4_F32 (A-operand and B-operand of the same data share the
// identical VGPR layout, so one float2 per lane feeds both).

typedef float v8f __attribute__((ext_vector_type(8)));
typedef float v2f __attribute__((ext_vector_type(2)));

#define BATCH      1024
#define FEAT       128
#define TILE_ROWS  16
#define NUM_TILES  (BATCH / TILE_ROWS)   // 64

__global__ __launch_bounds__(32)
void acl_tile_kernel(const float* __restrict__ x,
                     const int*   __restrict__ labels,
                     const float* __restrict__ centers,
                     const float* __restrict__ margin,
                     float*       __restrict__ partials)
{
    const int lane = threadIdx.x;          // wave32: 0..31
    const int tile = blockIdx.x;           // 0..63
    const int m    = lane & 15;            // matrix row handled by this lane
    const int half = lane >> 4;            // 0: K = kk,kk+1   1: K = kk+2,kk+3
    const int row  = tile * TILE_ROWS + m; // batch row
    const int koff = half * 2;

    // Gather: center row selected by this batch row's label.
    const float* xr = x       + (size_t)row          * FEAT + koff;
    const float* cr = centers + (size_t)labels[row]  * FEAT + koff;

    // Chain 32 WMMAs over K=128. acc diag accumulates ||x_row - c_row||^2.
    v8f acc = {};
#pragma unroll
    for (int kk = 0; kk < FEAT; kk += 4) {
        v2f xv = *(const v2f*)(xr + kk);   // 8B-aligned: kk%4==0, koff in {0,2}
        v2f cv = *(const v2f*)(cr + kk);
        v2f s;
        s.x = xv.x - cv.x;
        s.y = xv.y - cv.y;
        // D = A*B + C with A = S-slice (16x4), B = S^T-slice (4x16): same regs.
        acc = __builtin_amdgcn_wmma_f32_16x16x4_f32(
                  /*neg_a=*/false, s, /*neg_b=*/false, s,
                  /*c_mod=*/(short)0, acc, /*reuse_a=*/false, /*reuse_b=*/false);
    }

    // Diagonal extraction from the 16x16 f32 C/D layout:
    //   rows 0..7  -> lane m,     VGPR m
    //   rows 8..15 -> lane 16+m,  VGPR m-8
    float dist = 0.0f;
    if (lane < 8)        dist = acc[lane];
    else if (lane >= 24) dist = acc[lane - 24];

    float v = dist - margin[0];
    v = (v > 0.0f) ? v : 0.0f;
    if (lane >= 8 && lane < 24) v = 0.0f;  // lanes not holding a diag element

    // Wave32 butterfly reduction of the 16 live values.
#pragma unroll
    for (int off = 16; off >= 1; off >>= 1)
        v += __shfl_xor(v, off, 32);

    if (lane == 0) partials[tile] = v;
}

__global__ __launch_bounds__(32)
void acl_reduce_kernel(const float* __restrict__ partials,
                       float*       __restrict__ out)
{
    const int lane = threadIdx.x;          // 32 lanes, 64 partials
    float v = partials[lane] + partials[lane + 32];
#pragma unroll
    for (int off = 16; off >= 1; off >>= 1)
        v += __shfl_xor(v, off, 32);
    if (lane == 0) out[0] = v * (1.0f / (float)BATCH);
}

extern "C" void kernel_launch(void* const* d_in, const int* in_sizes, int n_in,
                              void* d_out, int out_size, void* d_ws, size_t ws_size,
                              hipStream_t stream)
{
    const float* x       = (const float*)d_in[0];  // (1024, 128) f32
    const int*   labels  = (const int*)  d_in[1];  // (1024,) int
    const float* centers = (const float*)d_in[2];  // (100000, 128) f32
    const float* margin  = (const float*)d_in[3];  // (1,) f32
    float*       out     = (float*)d_out;          // scalar loss
    float*       partials = (float*)d_ws;          // 64 floats of scratch

    // 64 single-wave workgroups: spreads the gather across WGPs, wave32 each,
    // uniform EXEC (WMMA requirement), no partial tiles (1024 % 16 == 0).
    acl_tile_kernel<<<NUM_TILES, 32, 0, stream>>>(x, labels, centers, margin, partials);
    acl_reduce_kernel<<<1, 32, 0, stream>>>(partials, out);
}